// VSSM_84645215470329
// MI455X (gfx1250) — compile-verified
//
#include <hip/hip_runtime.h>
#include <hip/hip_bf16.h>
#include <math.h>
#include <stdint.h>

typedef __attribute__((ext_vector_type(2))) float v2f;
typedef __attribute__((ext_vector_type(8))) float v8f;

#define B_      8
#define Hh      48
#define Ww      48
#define Lh      (Hh * Ww)        // 2304
#define DMODEL  192
#define DINNER  384
#define DSTATE  16
#define KDIR    4
#define DTRANK  12
#define CDBL    (DTRANK + 2 * DSTATE)  // 44
#define M_ROWS  (B_ * Lh)        // 18432

__device__ __forceinline__ float silu_(float x) {
    return x * (1.0f / (1.0f + __expf(-x)));
}

// scan-position -> row-major-position map for the 4 directions
__device__ __forceinline__ int dirmap(int k, int p) {
    switch (k) {
        case 0:  return p;
        case 1:  return (p % Hh) * Ww + (p / Hh);
        case 2:  return Lh - 1 - p;
        default: { int q = Lh - 1 - p; return (q % Hh) * Ww + (q / Hh); }
    }
}

// ---------------------------------------------------------------------------
// C[M,N] = A[M,Kd] * W[N,Kd]^T   (fp32 WMMA 16x16x4)
// One wave -> 16(M) x 64(N). A fragment reused across 4 accumulators and the
// next k-step's fragments are register-prefetched while the current WMMAs run.
// ---------------------------------------------------------------------------
__global__ __launch_bounds__(32) void gemm_abT_wmma(
    const float* __restrict__ A, const float* __restrict__ W,
    float* __restrict__ C, int N, int Kd)
{
    const int lane = threadIdx.x;
    const int half = lane >> 4;      // 0: lanes 0-15, 1: lanes 16-31
    const int lr   = lane & 15;
    const int n0 = blockIdx.x * 64;
    const int m0 = blockIdx.y * 16;
    const float* Ap = A + (size_t)(m0 + lr) * Kd;
    const float* Wp[4];
#pragma unroll
    for (int j = 0; j < 4; ++j) Wp[j] = W + (size_t)(n0 + 16 * j + lr) * Kd;

    v8f acc[4] = {{}, {}, {}, {}};
    v2f a_cur, b_cur[4];
    {   // prime the pipeline
        const int kk = 2 * half;
        a_cur.x = Ap[kk]; a_cur.y = Ap[kk + 1];
#pragma unroll
        for (int j = 0; j < 4; ++j) { b_cur[j].x = Wp[j][kk]; b_cur[j].y = Wp[j][kk + 1]; }
    }
    for (int k0 = 0; k0 < Kd; k0 += 4) {
        v2f a_nxt = {}, b_nxt[4] = {{}, {}, {}, {}};
        if (k0 + 4 < Kd) {           // issue next step's loads before computing
            const int kn = k0 + 4 + 2 * half;
            a_nxt.x = Ap[kn]; a_nxt.y = Ap[kn + 1];
#pragma unroll
            for (int j = 0; j < 4; ++j) { b_nxt[j].x = Wp[j][kn]; b_nxt[j].y = Wp[j][kn + 1]; }
        }
#pragma unroll
        for (int j = 0; j < 4; ++j)
            acc[j] = __builtin_amdgcn_wmma_f32_16x16x4_f32(
                         false, a_cur, false, b_cur[j], (short)0, acc[j], false, false);
        a_cur = a_nxt;
#pragma unroll
        for (int j = 0; j < 4; ++j) b_cur[j] = b_nxt[j];
    }
#pragma unroll
    for (int j = 0; j < 4; ++j)
#pragma unroll
        for (int r = 0; r < 8; ++r)
            C[(size_t)(m0 + r + 8 * half) * N + n0 + 16 * j + lr] = acc[j][r];
}

// ---------------------------------------------------------------------------
// Depthwise 3x3 conv (SAME, zero pad) + bias + SiLU.
// Reads xi half of xz (b,l,768) cols [0,384); writes xconv as (b, d, l).
// ---------------------------------------------------------------------------
__global__ void conv_silu_kernel(
    const float* __restrict__ xz, const float* __restrict__ cw,
    const float* __restrict__ cb, float* __restrict__ xconv)
{
    int idx = blockIdx.x * blockDim.x + threadIdx.x;
    if (idx >= B_ * DINNER * Lh) return;
    const int l = idx % Lh;
    const int d = (idx / Lh) % DINNER;
    const int b = idx / (Lh * DINNER);
    const int h = l / Ww, w = l % Ww;
    float acc = cb[d];
#pragma unroll
    for (int ky = -1; ky <= 1; ++ky)
#pragma unroll
        for (int kx = -1; kx <= 1; ++kx) {
            const int hh = h + ky, ww = w + kx;
            if (hh >= 0 && hh < Hh && ww >= 0 && ww < Ww)
                acc += xz[((size_t)b * Lh + hh * Ww + ww) * (2 * DINNER) + d]
                     * cw[d * 9 + (ky + 1) * 3 + (kx + 1)];
        }
    xconv[idx] = silu_(acc);
}

// ---------------------------------------------------------------------------
// x_dbl[b,k,c,l] = sum_d xs[b,k,d,l] * x_proj_w[k,c,d]
// xs gathered on the fly via dirmap. Wave tile: 16(c) x 64(l). Pipelined.
// ---------------------------------------------------------------------------
__global__ __launch_bounds__(32) void gemm_xproj_wmma(
    const float* __restrict__ xconv, const float* __restrict__ xpw,
    float* __restrict__ xdbl)
{
    const int lane = threadIdx.x, half = lane >> 4, lr = lane & 15;
    const int n0 = blockIdx.x * 64;
    const int m0 = blockIdx.y * 16;
    const int bk = blockIdx.z;
    const int b = bk / KDIR, k = bk % KDIR;
    int gcol[4];
#pragma unroll
    for (int j = 0; j < 4; ++j) gcol[j] = dirmap(k, n0 + 16 * j + lr);
    const float* xc = xconv + (size_t)b * DINNER * Lh;
    const float* Wk = xpw + (size_t)k * CDBL * DINNER;
    const int crow = m0 + lr;
    const bool arow_ok = (crow < CDBL);

    v8f acc[4] = {{}, {}, {}, {}};
    v2f a_cur, b_cur[4];
    {
        const int kk = 2 * half;
        a_cur.x = arow_ok ? Wk[(size_t)crow * DINNER + kk]     : 0.0f;
        a_cur.y = arow_ok ? Wk[(size_t)crow * DINNER + kk + 1] : 0.0f;
#pragma unroll
        for (int j = 0; j < 4; ++j) {
            b_cur[j].x = xc[(size_t)kk * Lh + gcol[j]];
            b_cur[j].y = xc[(size_t)(kk + 1) * Lh + gcol[j]];
        }
    }
    for (int k0 = 0; k0 < DINNER; k0 += 4) {
        v2f a_nxt = {}, b_nxt[4] = {{}, {}, {}, {}};
        if (k0 + 4 < DINNER) {
            const int kn = k0 + 4 + 2 * half;
            a_nxt.x = arow_ok ? Wk[(size_t)crow * DINNER + kn]     : 0.0f;
            a_nxt.y = arow_ok ? Wk[(size_t)crow * DINNER + kn + 1] : 0.0f;
#pragma unroll
            for (int j = 0; j < 4; ++j) {
                b_nxt[j].x = xc[(size_t)kn * Lh + gcol[j]];
                b_nxt[j].y = xc[(size_t)(kn + 1) * Lh + gcol[j]];
            }
        }
#pragma unroll
        for (int j = 0; j < 4; ++j)
            acc[j] = __builtin_amdgcn_wmma_f32_16x16x4_f32(
                         false, a_cur, false, b_cur[j], (short)0, acc[j], false, false);
        a_cur = a_nxt;
#pragma unroll
        for (int j = 0; j < 4; ++j) b_cur[j] = b_nxt[j];
    }
    float* outb = xdbl + (size_t)bk * CDBL * Lh;
#pragma unroll
    for (int j = 0; j < 4; ++j)
#pragma unroll
        for (int r = 0; r < 8; ++r) {
            const int row = m0 + r + 8 * half;
            if (row < CDBL) outb[(size_t)row * Lh + n0 + 16 * j + lr] = acc[j][r];
        }
}

// ---------------------------------------------------------------------------
// dt[b,k,d,l] = softplus( sum_r x_dbl[b,k,r,l] * dt_w[k,d,r] + dt_bias[k,d] )
// Kd = DT_RANK = 12 -> exactly 3 WMMA k-steps. Wave tile: 16(d) x 64(l).
// ---------------------------------------------------------------------------
__global__ __launch_bounds__(32) void gemm_dt_wmma(
    const float* __restrict__ xdbl, const float* __restrict__ dtw,
    const float* __restrict__ dtb, float* __restrict__ dtbuf)
{
    const int lane = threadIdx.x, half = lane >> 4, lr = lane & 15;
    const int n0 = blockIdx.x * 64;
    const int m0 = blockIdx.y * 16;
    const int bk = blockIdx.z;
    const int k = bk % KDIR;
    const float* Wk = dtw + (size_t)k * DINNER * DTRANK;
    const float* Xb = xdbl + (size_t)bk * CDBL * Lh;
    const int drow = m0 + lr;

    v8f acc[4] = {{}, {}, {}, {}};
    v2f a_cur, b_cur[4];
    {
        const int kk = 2 * half;
        a_cur.x = Wk[(size_t)drow * DTRANK + kk];
        a_cur.y = Wk[(size_t)drow * DTRANK + kk + 1];
#pragma unroll
        for (int j = 0; j < 4; ++j) {
            b_cur[j].x = Xb[(size_t)kk * Lh + n0 + 16 * j + lr];
            b_cur[j].y = Xb[(size_t)(kk + 1) * Lh + n0 + 16 * j + lr];
        }
    }
#pragma unroll
    for (int k0 = 0; k0 < DTRANK; k0 += 4) {
        v2f a_nxt = {}, b_nxt[4] = {{}, {}, {}, {}};
        if (k0 + 4 < DTRANK) {
            const int kn = k0 + 4 + 2 * half;
            a_nxt.x = Wk[(size_t)drow * DTRANK + kn];
            a_nxt.y = Wk[(size_t)drow * DTRANK + kn + 1];
#pragma unroll
            for (int j = 0; j < 4; ++j) {
                b_nxt[j].x = Xb[(size_t)kn * Lh + n0 + 16 * j + lr];
                b_nxt[j].y = Xb[(size_t)(kn + 1) * Lh + n0 + 16 * j + lr];
            }
        }
#pragma unroll
        for (int j = 0; j < 4; ++j)
            acc[j] = __builtin_amdgcn_wmma_f32_16x16x4_f32(
                         false, a_cur, false, b_cur[j], (short)0, acc[j], false, false);
        a_cur = a_nxt;
#pragma unroll
        for (int j = 0; j < 4; ++j) b_cur[j] = b_nxt[j];
    }
#pragma unroll
    for (int j = 0; j < 4; ++j)
#pragma unroll
        for (int r = 0; r < 8; ++r) {
            const int row = m0 + r + 8 * half;
            const int col = n0 + 16 * j + lr;
            float v = acc[j][r] + dtb[k * DINNER + row];
            float sp = (v > 20.0f) ? v : log1pf(__expf(v));   // softplus
            dtbuf[((size_t)bk * DINNER + row) * Lh + col] = sp;
        }
}

// ---------------------------------------------------------------------------
// Selective scan: one block per (b,k); thread d holds h[16] in registers.
// B_t/C_t staged through a double-buffered LDS slot. Next step's 32 values
// are fetched with global_load_async_to_lds_b32 (ASYNCcnt) overlapping the
// recurrence math; one barrier per step.
// ---------------------------------------------------------------------------
__global__ __launch_bounds__(DINNER) void scan_kernel(
    const float* __restrict__ xconv, const float* __restrict__ dtbuf,
    const float* __restrict__ xdbl, const float* __restrict__ A_logs,
    const float* __restrict__ Ds, float* __restrict__ ys)
{
    const int bk = blockIdx.x;
    const int b = bk / KDIR, k = bk % KDIR;
    const int d = threadIdx.x;
    __shared__ float sBC[2][2 * DSTATE];   // [buf][0,16): B_t, [16,32): C_t

    float An[DSTATE], hst[DSTATE];
#pragma unroll
    for (int n = 0; n < DSTATE; ++n) {
        An[n]  = -__expf(A_logs[((size_t)(k * DINNER + d)) * DSTATE + n]);
        hst[n] = 0.0f;
    }
    const float Dd = Ds[k * DINNER + d];
    const float* xc  = xconv + ((size_t)b * DINNER + d) * Lh;
    const float* dtp = dtbuf + ((size_t)bk * DINNER + d) * Lh;
    const float* bc  = xdbl  + (size_t)bk * CDBL * Lh;
    float* yo = ys + ((size_t)bk * DINNER + d) * Lh;

    // prologue: stage step 0
    if (d < 2 * DSTATE)
        sBC[0][d] = bc[(size_t)(DTRANK + d) * Lh + 0];   // rows 12..43 = B then C
    __syncthreads();

    for (int p = 0; p < Lh; ++p) {
        const int cur = p & 1;
        // async-prefetch next step's B/C directly into the other LDS buffer
        if ((p + 1) < Lh && d < 2 * DSTATE) {
            unsigned ldsa = (unsigned)(uintptr_t)&sBC[cur ^ 1][d];
            unsigned long long ga =
                (unsigned long long)(uintptr_t)&bc[(size_t)(DTRANK + d) * Lh + p + 1];
            asm volatile("global_load_async_to_lds_b32 %0, %1, off"
                         :: "v"(ldsa), "v"(ga) : "memory");
        }
        if ((p & 127) == 0)
            __builtin_prefetch(dtp + p + 128, 0, 0);

        const float dt = dtp[p];
        const float u  = xc[dirmap(k, p)];
        const float du = dt * u;
        float y = 0.0f;
#pragma unroll
        for (int n = 0; n < DSTATE; ++n) {
            hst[n] = __expf(dt * An[n]) * hst[n] + du * sBC[cur][n];
            y += hst[n] * sBC[cur][DSTATE + n];
        }
        yo[p] = y + Dd * u;

        asm volatile("s_wait_asynccnt 0" ::: "memory");   // drain async fill
        __syncthreads();
    }
}

// ---------------------------------------------------------------------------
// Merge 4 directions back to row-major, LayerNorm over D_INNER, gate by
// silu(z). One block per (b,l); 384 threads.
// ---------------------------------------------------------------------------
__global__ __launch_bounds__(DINNER) void merge_ln_gate_kernel(
    const float* __restrict__ ys, const float* __restrict__ xz,
    const float* __restrict__ lnw, const float* __restrict__ lnb,
    float* __restrict__ yfin)
{
    const int bl = blockIdx.x;
    const int b = bl / Lh, l = bl % Lh;
    const int d = threadIdx.x;
    const int h = l / Ww, w = l % Ww;
    const int t = w * Hh + h;                 // col-major position

    float y = ys[(((size_t)(b * KDIR + 0) * DINNER) + d) * Lh + l]
            + ys[(((size_t)(b * KDIR + 1) * DINNER) + d) * Lh + t]
            + ys[(((size_t)(b * KDIR + 2) * DINNER) + d) * Lh + (Lh - 1 - l)]
            + ys[(((size_t)(b * KDIR + 3) * DINNER) + d) * Lh + (Lh - 1 - t)];

    __shared__ float rsum[DINNER], rsq[DINNER];
    rsum[d] = y; rsq[d] = y * y;
    __syncthreads();
    for (int s = 256; s > 0; s >>= 1) {
        if (d < s && d + s < DINNER) { rsum[d] += rsum[d + s]; rsq[d] += rsq[d + s]; }
        __syncthreads();
    }
    const float mu  = rsum[0] * (1.0f / DINNER);
    const float var = rsq[0] * (1.0f / DINNER) - mu * mu;
    const float yn  = (y - mu) * rsqrtf(var + 1e-6f) * lnw[d] + lnb[d];
    const float zv  = xz[(size_t)bl * (2 * DINNER) + DINNER + d];
    yfin[(size_t)bl * DINNER + d] = yn * silu_(zv);
}

// ---------------------------------------------------------------------------
extern "C" void kernel_launch(void* const* d_in, const int* in_sizes, int n_in,
                              void* d_out, int out_size, void* d_ws, size_t ws_size,
                              hipStream_t stream)
{
    const float* x         = (const float*)d_in[0];
    const float* in_proj_w = (const float*)d_in[1];
    const float* conv_w    = (const float*)d_in[2];
    const float* conv_b    = (const float*)d_in[3];
    const float* x_proj_w  = (const float*)d_in[4];
    const float* dt_w      = (const float*)d_in[5];
    const float* dt_bias   = (const float*)d_in[6];
    const float* A_logs    = (const float*)d_in[7];
    const float* Ds        = (const float*)d_in[8];
    const float* ln_w      = (const float*)d_in[9];
    const float* ln_b      = (const float*)d_in[10];
    const float* out_proj_w= (const float*)d_in[11];
    float* out = (float*)d_out;

    // workspace carve-up (floats)
    float* W = (float*)d_ws;
    size_t off = 0;
    float* xz    = W + off; off += (size_t)M_ROWS * 2 * DINNER;       // 14.2M
    float* xconv = W + off; off += (size_t)B_ * DINNER * Lh;          //  7.1M
    float* xdbl  = W + off; off += (size_t)B_ * KDIR * CDBL * Lh;     //  3.2M
    float* dtbuf = W + off; off += (size_t)B_ * KDIR * DINNER * Lh;   // 28.3M
    float* ysbuf = W + off; off += (size_t)B_ * KDIR * DINNER * Lh;   // 28.3M
    float* yfin  = W + off; off += (size_t)M_ROWS * DINNER;           //  7.1M

    // 1) xz = x @ in_proj_w^T   (18432 x 768, K=192)
    gemm_abT_wmma<<<dim3((2 * DINNER) / 64, M_ROWS / 16), 32, 0, stream>>>(
        x, in_proj_w, xz, 2 * DINNER, DMODEL);

    // 2) depthwise conv + silu -> xconv (b,d,l)
    conv_silu_kernel<<<(B_ * DINNER * Lh + 255) / 256, 256, 0, stream>>>(
        xz, conv_w, conv_b, xconv);

    // 3) x_dbl = x_proj_w @ xs  (44 x 2304 per (b,k), K=384, gathered B)
    gemm_xproj_wmma<<<dim3(Lh / 64, (CDBL + 15) / 16, B_ * KDIR), 32, 0, stream>>>(
        xconv, x_proj_w, xdbl);

    // 4) dt = softplus(dt_w @ dts + bias)  (384 x 2304 per (b,k), K=12)
    gemm_dt_wmma<<<dim3(Lh / 64, DINNER / 16, B_ * KDIR), 32, 0, stream>>>(
        xdbl, dt_w, dt_bias, dtbuf);

    // 5) selective scan (serial critical path; async LDS double-buffering)
    scan_kernel<<<B_ * KDIR, DINNER, 0, stream>>>(
        xconv, dtbuf, xdbl, A_logs, Ds, ysbuf);

    // 6) merge directions + LayerNorm + silu(z) gate
    merge_ln_gate_kernel<<<B_ * Lh, DINNER, 0, stream>>>(
        ysbuf, xz, ln_w, ln_b, yfin);

    // 7) out = yfin @ out_proj_w^T  (18432 x 192, K=384)
    gemm_abT_wmma<<<dim3(DMODEL / 64, M_ROWS / 16), 32, 0, stream>>>(
        yfin, out_proj_w, out, DMODEL, DINNER);
}